// CrinaSynapse_31550829756815
// MI455X (gfx1250) — compile-verified
//
#include <hip/hip_runtime.h>

#define D_MODEL   1024
#define N_HEADS   16
#define HEAD_DIM  64
#define NUM_NODES 15
#define BATCH     8
#define SEQ_Q     4096

typedef __attribute__((ext_vector_type(2))) float v2f;
typedef __attribute__((ext_vector_type(4))) float f4;
typedef __attribute__((ext_vector_type(8))) float v8f;

// One 16x16 (M=16 pairs, N=16 outputs) tile of  out = A(16x64) * W^T  using
// V_WMMA_F32_16X16X4_F32, accumulating over 16 K-steps of 4.
// A-layout (ISA 16x4 f32): lane c (0..15) / c+16 holds K = kk*4 + 2*kh + {0,1}
// B-layout (4x16): reg0/reg1 hold rows kk*4+2*kh+{0,1}, N striped across lanes.
// B[d][e] = W[e*64 + d], so each lane loads a contiguous float2 of W's row e0+c.
__device__ inline v8f wmma_node_gemm(const float* __restrict__ fA,     // LDS, 16x64 row-major (m,d)
                                     const float* __restrict__ Wnode,  // global, 64x64 row-major (e,d)
                                     int e0, int c, int kh) {
  v8f acc = {0.f, 0.f, 0.f, 0.f, 0.f, 0.f, 0.f, 0.f};
#pragma unroll
  for (int kk = 0; kk < 16; ++kk) {
    const int dbase = kk * 4 + 2 * kh;
    v2f a = *(const v2f*)(fA + c * 64 + dbase);
    v2f b = *(const v2f*)(Wnode + (e0 + c) * 64 + dbase);
    acc = __builtin_amdgcn_wmma_f32_16x16x4_f32(false, a, false, b, (short)0, acc,
                                                false, false);
  }
  return acc;
}

// Stage 1: binary-tree of node GEMMs + LIF gating, batched over 16 (b,h) pairs
// per workgroup (8 workgroups x 16 = 128 pairs). Emits addv[b][h*64+e] = tree_v.
__global__ __launch_bounds__(256) void tree_kernel(
    const float* __restrict__ key_value, const float* __restrict__ W,
    const float* __restrict__ bias, const float* __restrict__ lif_th,
    const float* __restrict__ lif_tau, const float* __restrict__ lif_vreset,
    float* __restrict__ addv) {
  extern __shared__ float smem[];
  float* okb  = smem;          // [8 nodes][16 m][64]  = 8192 f
  float* ovb  = smem + 8192;   // [8 nodes][16 m][64]  = 8192 f
  float* xbuf = smem + 16384;  // leaf x [8][16][64]   = 8192 f (reused below)
  float* fk   = xbuf;          // [4 nodes][16][64]    = 4096 f
  float* fv   = xbuf + 4096;   // [4 nodes][16][64]    = 4096 f

  const int tid  = threadIdx.x;
  const int g    = blockIdx.x;      // pair group: pairs p = g*16 .. g*16+15
  const int lane = tid & 31;
  const int w    = tid >> 5;        // wave id 0..7
  const int c    = lane & 15;
  const int kh   = lane >> 4;

  // ---- load leaf inputs x[node][m][d] = kv[b, node, h, d] ----
  for (int t = tid; t < 8192; t += 256) {
    const int node = t >> 10, rem = t & 1023, m = rem >> 6, d = rem & 63;
    const int p = g * 16 + m, b = p >> 4, h = p & 15;
    xbuf[t] = key_value[b * 16384 + node * 1024 + h * 64 + d];
  }
  __syncthreads();

  // ---- leaf level: wave w owns node w (weight index 7+w), fk == fv == x ----
  {
    const int wi = 7 + w;
    const float* fA = xbuf + w * 1024;
#pragma unroll
    for (int nt = 0; nt < 4; ++nt) {
      const int e0 = nt * 16;
      v8f acc = wmma_node_gemm(fA, W + wi * 4096, e0, c, kh);
      const int ei = wi * 64 + e0 + c;
      const float be = bias[ei], th = lif_th[ei], tau = lif_tau[ei], vr = lif_vreset[ei];
#pragma unroll
      for (int r = 0; r < 8; ++r) {
        const int m = r + 8 * kh;
        const int off = w * 1024 + m * 64 + e0 + c;
        const float xe  = xbuf[off];
        const float val = acc[r] + be;
        const bool  sk  = xe >= th;
        const float s1  = sk ? vr : xe;
        const float v2  = tau * s1 + xe;
        const bool  sv  = v2 >= th;
        okb[off] = sk ? val : 0.f;
        ovb[off] = sv ? val : 0.f;
      }
    }
  }
  __syncthreads();  // also fences xbuf before its reuse as fk/fv

  // ---- inner levels: size 4 -> 2 -> 1, wave pair (k-path, v-path) per node ----
  for (int size = 4; size >= 1; size >>= 1) {
    const int start = size - 1;
    const int nact  = 2 * size;
    const int node  = w >> 1, path = w & 1;
    if (w < nact) {  // wave-uniform: EXEC stays all-ones for WMMA
      const float* src = (path == 0) ? okb : ovb;
      float* dst = ((path == 0) ? fk : fv) + node * 1024;
      for (int t = lane; t < 1024; t += 32)
        dst[t] = 0.5f * (src[(2 * node) * 1024 + t] + src[(2 * node + 1) * 1024 + t]);
    }
    __syncthreads();
    if (w < nact) {
      const int wi = start + node;
      const float* fA = ((path == 0) ? fk : fv) + node * 1024;
#pragma unroll
      for (int nt = 0; nt < 4; ++nt) {
        const int e0 = nt * 16;
        v8f acc = wmma_node_gemm(fA, W + wi * 4096, e0, c, kh);
        const int ei = wi * 64 + e0 + c;
        const float be = bias[ei], th = lif_th[ei], tau = lif_tau[ei], vr = lif_vreset[ei];
#pragma unroll
        for (int r = 0; r < 8; ++r) {
          const int m   = r + 8 * kh;
          const int off = node * 1024 + m * 64 + e0 + c;
          const float val = acc[r] + be;
          const float fkv = fk[off];
          const bool  sk  = fkv >= th;
          if (path == 0) {
            okb[off] = sk ? val : 0.f;
          } else {
            const float fvv = fv[off];
            const float s1  = sk ? vr : fkv;
            const float v2  = tau * s1 + fvv;
            const bool  sv  = v2 >= th;
            ovb[off] = sv ? val : 0.f;
          }
        }
      }
    }
    __syncthreads();
  }

  // ---- root ov == tree_v; softmax over singleton axis == 1, so this is the
  //      entire attention output. Flatten heads into addv[b][1024]. ----
  for (int t = tid; t < 1024; t += 256) {
    const int m = t >> 6, e = t & 63;
    const int p = g * 16 + m, b = p >> 4, h = p & 15;
    addv[b * 1024 + h * 64 + e] = ovb[t];
  }
}

// Stage 2 (bandwidth-bound): out = LN(query + addv[b]) * gamma + beta.
// One block per 1024-element row; float4 per thread; NT load/store (stream once).
__global__ __launch_bounds__(256) void add_ln_kernel(
    const float* __restrict__ query, const float* __restrict__ addv,
    const float* __restrict__ gamma, const float* __restrict__ beta,
    float* __restrict__ out) {
  const int row  = blockIdx.x;
  const int b    = row >> 12;  // row / SEQ_Q
  const int t    = threadIdx.x;
  const int lane = t & 31, wv = t >> 5;

  const f4 q = __builtin_nontemporal_load((const f4*)(query + (size_t)row * 1024) + t);
  const f4 a = ((const f4*)(addv + b * 1024))[t];
  const f4 r = q + a;

  float sum = r.x + r.y + r.z + r.w;
  float sq  = r.x * r.x + r.y * r.y + r.z * r.z + r.w * r.w;
#pragma unroll
  for (int off = 16; off > 0; off >>= 1) {
    sum += __shfl_down(sum, off, 32);
    sq  += __shfl_down(sq, off, 32);
  }
  __shared__ float s_sum[8], s_sq[8];
  if (lane == 0) { s_sum[wv] = sum; s_sq[wv] = sq; }
  __syncthreads();
  float tot = 0.f, totq = 0.f;
#pragma unroll
  for (int i = 0; i < 8; ++i) { tot += s_sum[i]; totq += s_sq[i]; }

  const float mean = tot * (1.0f / 1024.0f);
  const float var  = totq * (1.0f / 1024.0f) - mean * mean;
  const float inv  = rsqrtf(var + 1e-5f);

  const f4 gm = ((const f4*)gamma)[t];
  const f4 bt = ((const f4*)beta)[t];
  f4 o;
  o.x = (r.x - mean) * inv * gm.x + bt.x;
  o.y = (r.y - mean) * inv * gm.y + bt.y;
  o.z = (r.z - mean) * inv * gm.z + bt.z;
  o.w = (r.w - mean) * inv * gm.w + bt.w;
  __builtin_nontemporal_store(o, (f4*)(out + (size_t)row * 1024) + t);
}

extern "C" void kernel_launch(void* const* d_in, const int* in_sizes, int n_in,
                              void* d_out, int out_size, void* d_ws, size_t ws_size,
                              hipStream_t stream) {
  const float* query      = (const float*)d_in[0];
  const float* key_value  = (const float*)d_in[1];
  const float* W          = (const float*)d_in[2];
  const float* bias       = (const float*)d_in[3];
  const float* lif_th     = (const float*)d_in[4];
  const float* lif_tau    = (const float*)d_in[5];
  const float* lif_vreset = (const float*)d_in[6];
  const float* ln_gamma   = (const float*)d_in[7];
  const float* ln_beta    = (const float*)d_in[8];
  float* out  = (float*)d_out;
  float* addv = (float*)d_ws;  // BATCH * 1024 floats of scratch

  const size_t smem = 24576 * sizeof(float);  // 96 KB (<< 320 KB WGP LDS)
  tree_kernel<<<dim3(8), dim3(256), smem, stream>>>(key_value, W, bias, lif_th,
                                                    lif_tau, lif_vreset, addv);
  add_ln_kernel<<<dim3(BATCH * SEQ_Q), dim3(256), 0, stream>>>(query, addv,
                                                               ln_gamma, ln_beta, out);
}